// diffRNN_75591424410018
// MI455X (gfx1250) — compile-verified
//
#include <hip/hip_runtime.h>
#include <hip/hip_bf16.h>
#include <math.h>

// Dims from the reference
#define BB 64
#define VV 50
#define CC 20
#define DD 256
#define TDIFF 1000

typedef __attribute__((ext_vector_type(2))) float v2f;
typedef __attribute__((ext_vector_type(8))) float v8f;

__device__ __forceinline__ float sigmoidf(float x) { return 1.0f / (1.0f + __expf(-x)); }

// ---------------------------------------------------------------------------
// K0: per-batch diffusion constants + sin/cos timestep embedding.
// grid = B, block = 256
// ---------------------------------------------------------------------------
__global__ void k_precompute(const int* __restrict__ tdiff, float* __restrict__ sqa,
                             float* __restrict__ sq1ma, float* __restrict__ temb) {
  int b = blockIdx.x;
  int t = tdiff[b];
  if (threadIdx.x == 0) {
    float prod = 1.0f;
    for (int i = 0; i <= t; ++i) {
      float beta = 1e-4f + (0.02f - 1e-4f) * (float)i / (float)(TDIFF - 1);
      prod *= (1.0f - beta);
    }
    sqa[b] = sqrtf(prod);
    sq1ma[b] = sqrtf(1.0f - prod);
  }
  int d = threadIdx.x;
  float tf = (float)t;
  float lg = logf(10000.0f);
  if (d < 128) {
    float freq = __expf(-lg * (float)d / 128.0f);
    temb[b * DD + d] = sinf(tf * freq);
  } else {
    float freq = __expf(-lg * (float)(d - 128) / 128.0f);
    temb[b * DD + d] = cosf(tf * freq);
  }
}

// ---------------------------------------------------------------------------
// K1: visit_embedding = relu(emb[seqs]).sum(C) + time_encoding; also builds
// the diff-MLP input (x_noisy + t_emb) and copies normal_noise to the output.
// grid = B*V, block = 256 (one d per thread)
// ---------------------------------------------------------------------------
__global__ void k_visit(const int* __restrict__ seqs, const float* __restrict__ stime,
                        const float* __restrict__ noise, const float* __restrict__ emb,
                        const float* __restrict__ tw1, const float* __restrict__ tb1,
                        const float* __restrict__ tw2, const float* __restrict__ tb2,
                        const float* __restrict__ sqa, const float* __restrict__ sq1ma,
                        const float* __restrict__ temb, float* __restrict__ ve,
                        float* __restrict__ xin, float* __restrict__ out_noise) {
  int bv = blockIdx.x;
  int b = bv / VV;
  __shared__ float tf[64];
  int d = threadIdx.x;
  float tval = stime[bv] / 180.0f;
  if (d < 64) {
    float u = tval * tw1[d] + tb1[d];
    float th = tanhf(u * u);
    tf[d] = 1.0f - th;
  }
  __syncthreads();
  float te = tb2[d];
#pragma unroll 8
  for (int k = 0; k < 64; ++k) te += tf[k] * tw2[d * 64 + k];
  float s = 0.0f;
  const int* srow = seqs + bv * CC;
  for (int c = 0; c < CC; ++c) {
    float e = emb[(size_t)srow[c] * DD + d];
    s += fmaxf(e, 0.0f);
  }
  float v = s + te;
  ve[(size_t)bv * DD + d] = v;
  float nz = noise[(size_t)bv * DD + d];
  out_noise[(size_t)bv * DD + d] = nz;
  xin[(size_t)bv * DD + d] = v * sqa[b] + nz * sq1ma[b] + temb[b * DD + d];
}

// ---------------------------------------------------------------------------
// WMMA f32 GEMM, register-tiled + software-pipelined.
// Each wave computes a 16x64 strip of C (4 n-tiles sharing one A fragment
// per K-step) via V_WMMA_F32_16X16X4_F32, double-buffering the fragment
// loads so the next K-step's loads are in flight while the current step's
// 4 WMMAs execute (critical for latency hiding at our low wave counts).
// Block = 256 (8 waves) covers a 32x256 macro-tile; grid = (M/32, N/256).
// TRANSB=1: B(k,n) = Bsrc[n*K + k] (x @ W.T).
// Requires M%32==0, N%256==0, K%4==0 (true for all calls here).
// ---------------------------------------------------------------------------
template <int TRANSB, int RELU>
__global__ void k_wmma_gemm(const float* __restrict__ A, const float* __restrict__ Bm,
                            const float* __restrict__ bias, float* __restrict__ C,
                            int M, int N, int K) {
  int wave = threadIdx.x >> 5;
  int lane = threadIdx.x & 31;
  int half = lane >> 4;    // 0/1 -> K pair {0,1} or {2,3}
  int idx16 = lane & 15;
  int tm = blockIdx.x * 2 + (wave >> 2);
  int col0 = blockIdx.y * 256 + (wave & 3) * 64 + idx16;  // first of 4 cols (+16 each)
  int row = tm * 16 + idx16;
  int kb = half * 2;

  const float* pA = A + (size_t)row * K + kb;
  const float* pB0;
  const float* pB1;
  const float* pB2;
  const float* pB3;
  const float* pBn = nullptr;  // non-transposed walker
  if (TRANSB) {
    pB0 = Bm + (size_t)(col0 +  0) * K + kb;
    pB1 = Bm + (size_t)(col0 + 16) * K + kb;
    pB2 = Bm + (size_t)(col0 + 32) * K + kb;
    pB3 = Bm + (size_t)(col0 + 48) * K + kb;
  } else {
    pBn = Bm + (size_t)kb * N + col0;
    pB0 = pB1 = pB2 = pB3 = nullptr;
  }

  // Stage 0: prologue loads for k=0
  v2f a, b0, b1, b2, b3;
  a = *(const v2f*)pA;  pA += 4;
  if (TRANSB) {
    b0 = *(const v2f*)pB0;  pB0 += 4;
    b1 = *(const v2f*)pB1;  pB1 += 4;
    b2 = *(const v2f*)pB2;  pB2 += 4;
    b3 = *(const v2f*)pB3;  pB3 += 4;
  } else {
    b0.x = pBn[0];   b0.y = pBn[N];
    b1.x = pBn[16];  b1.y = pBn[N + 16];
    b2.x = pBn[32];  b2.y = pBn[N + 32];
    b3.x = pBn[48];  b3.y = pBn[N + 48];
    pBn += (size_t)4 * N;
  }

  v8f acc0 = {}, acc1 = {}, acc2 = {}, acc3 = {};
  for (int k0 = 0; k0 < K - 4; k0 += 4) {
    // Issue next K-step's fragment loads before consuming the current ones.
    v2f an, b0n, b1n, b2n, b3n;
    an = *(const v2f*)pA;  pA += 4;
    if (TRANSB) {
      b0n = *(const v2f*)pB0;  pB0 += 4;
      b1n = *(const v2f*)pB1;  pB1 += 4;
      b2n = *(const v2f*)pB2;  pB2 += 4;
      b3n = *(const v2f*)pB3;  pB3 += 4;
    } else {
      b0n.x = pBn[0];   b0n.y = pBn[N];
      b1n.x = pBn[16];  b1n.y = pBn[N + 16];
      b2n.x = pBn[32];  b2n.y = pBn[N + 32];
      b3n.x = pBn[48];  b3n.y = pBn[N + 48];
      pBn += (size_t)4 * N;
    }
    acc0 = __builtin_amdgcn_wmma_f32_16x16x4_f32(false, a, false, b0, (short)0, acc0, false, false);
    acc1 = __builtin_amdgcn_wmma_f32_16x16x4_f32(false, a, false, b1, (short)0, acc1, false, false);
    acc2 = __builtin_amdgcn_wmma_f32_16x16x4_f32(false, a, false, b2, (short)0, acc2, false, false);
    acc3 = __builtin_amdgcn_wmma_f32_16x16x4_f32(false, a, false, b3, (short)0, acc3, false, false);
    a = an;  b0 = b0n;  b1 = b1n;  b2 = b2n;  b3 = b3n;
  }
  // Epilogue: consume the last prefetched fragments.
  acc0 = __builtin_amdgcn_wmma_f32_16x16x4_f32(false, a, false, b0, (short)0, acc0, false, false);
  acc1 = __builtin_amdgcn_wmma_f32_16x16x4_f32(false, a, false, b1, (short)0, acc1, false, false);
  acc2 = __builtin_amdgcn_wmma_f32_16x16x4_f32(false, a, false, b2, (short)0, acc2, false, false);
  acc3 = __builtin_amdgcn_wmma_f32_16x16x4_f32(false, a, false, b3, (short)0, acc3, false, false);

  int mbase = tm * 16 + half * 8;
  float* pC = C + (size_t)mbase * N + col0;
  float bv0 = bias ? bias[col0]      : 0.0f;
  float bv1 = bias ? bias[col0 + 16] : 0.0f;
  float bv2 = bias ? bias[col0 + 32] : 0.0f;
  float bv3 = bias ? bias[col0 + 48] : 0.0f;
#pragma unroll
  for (int i = 0; i < 8; ++i) {
    float o0 = acc0[i] + bv0;
    float o1 = acc1[i] + bv1;
    float o2 = acc2[i] + bv2;
    float o3 = acc3[i] + bv3;
    if (RELU) {
      o0 = fmaxf(o0, 0.0f); o1 = fmaxf(o1, 0.0f);
      o2 = fmaxf(o2, 0.0f); o3 = fmaxf(o3, 0.0f);
    }
    pC[0]  = o0;
    pC[16] = o1;
    pC[32] = o2;
    pC[48] = o3;
    pC += N;
  }
}

// gen1 = visit_embedding + (noise - predicted_noise)
__global__ void k_gen1(const float* __restrict__ ve, const float* __restrict__ noise,
                       const float* __restrict__ pred, float* __restrict__ gen1) {
  size_t i = (size_t)blockIdx.x * blockDim.x + threadIdx.x;
  gen1[i] = ve[i] + noise[i] - pred[i];
}

// xall[v][j][d]: j<64 -> ve[b=j][v], else gen2[b=j-64][v]   (V,128,256)
__global__ void k_pack(const float* __restrict__ ve, const float* __restrict__ gen2,
                       float* __restrict__ xall) {
  size_t i = (size_t)blockIdx.x * blockDim.x + threadIdx.x;
  int d = (int)(i & 255);
  int j = (int)((i >> 8) & 127);
  int v = (int)(i >> 15);
  float val = (j < BB) ? ve[((size_t)(j * VV + v)) * DD + d]
                       : gen2[((size_t)((j - BB) * VV + v)) * DD + d];
  xall[i] = val;
}

__global__ void k_zero(float* __restrict__ p, int n) {
  int i = blockIdx.x * blockDim.x + threadIdx.x;
  if (i < n) p[i] = 0.0f;
}

// LSTM cell pointwise for step v. grid = 128 (rows), block = 256 (d)
__global__ void k_cell(const float* __restrict__ XW, const float* __restrict__ ghh,
                       const float* __restrict__ bih, const float* __restrict__ bhh,
                       float* __restrict__ hcur, float* __restrict__ ccur,
                       float* __restrict__ hsall, int v) {
  int j = blockIdx.x;
  int d = threadIdx.x;
  const float* xrow = XW + (size_t)(v * 128 + j) * 1024;
  const float* grow = ghh + (size_t)j * 1024;
  float gi = xrow[d] + grow[d] + bih[d] + bhh[d];
  float gf = xrow[256 + d] + grow[256 + d] + bih[256 + d] + bhh[256 + d];
  float gg = xrow[512 + d] + grow[512 + d] + bih[512 + d] + bhh[512 + d];
  float go = xrow[768 + d] + grow[768 + d] + bih[768 + d] + bhh[768 + d];
  float c = sigmoidf(gf) * ccur[j * DD + d] + sigmoidf(gi) * tanhf(gg);
  float h = sigmoidf(go) * tanhf(c);
  ccur[j * DD + d] = c;
  hcur[j * DD + d] = h;
  hsall[(size_t)(j * VV + v) * DD + d] = h;
}

// Final classifier layer (N=2) + writes res / res_gen / last-step slices.
__global__ void k_clf3(const float* __restrict__ h2, const float* __restrict__ c3w,
                       const float* __restrict__ c3b, float* __restrict__ out) {
  int idx = blockIdx.x * blockDim.x + threadIdx.x;  // 12800 = 6400 rows * 2
  if (idx >= 6400 * 2) return;
  int o = idx & 1;
  int rowI = idx >> 1;  // j*V + v
  int j = rowI / VV;
  int v = rowI % VV;
  const float* h = h2 + (size_t)rowI * 512;
  const float* w = c3w + o * 512;
  float s = c3b[o];
  for (int k = 0; k < 512; ++k) s += h[k] * w[k];
  if (j < BB) {
    out[(j * VV + v) * 2 + o] = s;
    if (v == VV - 1) out[12800 + j * 2 + o] = s;
  } else {
    out[6400 + ((j - BB) * VV + v) * 2 + o] = s;
    if (v == VV - 1) out[12928 + (j - BB) * 2 + o] = s;
  }
}

extern "C" void kernel_launch(void* const* d_in, const int* in_sizes, int n_in,
                              void* d_out, int out_size, void* d_ws, size_t ws_size,
                              hipStream_t stream) {
  (void)in_sizes; (void)n_in; (void)out_size; (void)ws_size;
  const int*   seqs  = (const int*)d_in[0];
  const float* stime = (const float*)d_in[1];
  const int*   tdiff = (const int*)d_in[2];
  const float* noise = (const float*)d_in[3];
  const float* emb   = (const float*)d_in[4];
  const float* tw1   = (const float*)d_in[5];
  const float* tb1   = (const float*)d_in[6];
  const float* tw2   = (const float*)d_in[7];
  const float* tb2   = (const float*)d_in[8];
  const float* dw1   = (const float*)d_in[9];
  const float* db1   = (const float*)d_in[10];
  const float* dw2   = (const float*)d_in[11];
  const float* db2   = (const float*)d_in[12];
  const float* awv   = (const float*)d_in[13];
  const float* abv   = (const float*)d_in[14];
  const float* awo   = (const float*)d_in[15];
  const float* abo   = (const float*)d_in[16];
  const float* wih   = (const float*)d_in[17];
  const float* whh   = (const float*)d_in[18];
  const float* bih   = (const float*)d_in[19];
  const float* bhh   = (const float*)d_in[20];
  const float* c1w   = (const float*)d_in[21];
  const float* c1b   = (const float*)d_in[22];
  const float* c2w   = (const float*)d_in[23];
  const float* c2b   = (const float*)d_in[24];
  const float* c3w   = (const float*)d_in[25];
  const float* c3b   = (const float*)d_in[26];

  float* out = (float*)d_out;
  float* ws  = (float*)d_ws;

  // Workspace layout (floats); total 17,416,320 floats (~66.4 MB), with reuse.
  float* sqa   = ws;                  // 64
  float* sq1ma = ws + 64;             // 64
  float* temb  = ws + 128;            // 16384
  float* ve    = ws + 16512;          // 819200
  float* xin   = ws + 835712;         // 819200 (later: gen1)
  float* h1    = ws + 1654912;        // 1638400 (diff hidden; later attn tmp)
  float* gen2  = ws + 3293312;        // 819200
  float* xall  = ws + 4112512;        // 1638400
  float* XW    = ws + 5750912;        // 6553600 (lstm input proj; later clf h1)
  float* hsall = ws + 12304512;       // 1638400
  float* h2c   = ws + 13942912;       // 3276800
  float* hcur  = ws + 17219712;       // 32768
  float* ccur  = ws + 17252480;       // 32768
  float* gates = ws + 17285248;       // 131072

  // Output slots (floats): res@0(6400) resg@6400 resl@12800(128) resgl@12928(128)
  //                        noise@13056(819200) pred@832256(819200)
  float* out_noise = out + 13056;
  float* out_pred  = out + 832256;

  k_precompute<<<BB, 256, 0, stream>>>(tdiff, sqa, sq1ma, temb);
  k_visit<<<BB * VV, 256, 0, stream>>>(seqs, stime, noise, emb, tw1, tb1, tw2, tb2,
                                       sqa, sq1ma, temb, ve, xin, out_noise);

  // diff MLP: h = relu(xin @ dw1 + db1); pred = h @ dw2 + db2
  k_wmma_gemm<0, 1><<<dim3(100, 2), 256, 0, stream>>>(xin, dw1, db1, h1, 3200, 512, 256);
  k_wmma_gemm<0, 0><<<dim3(100, 1), 256, 0, stream>>>(h1, dw2, db2, out_pred, 3200, 256, 512);

  k_gen1<<<3200, 256, 0, stream>>>(ve, noise, out_pred, xin);  // gen1 in xin

  // attention projections: gen2 = (gen1 @ wv.T + bv) @ wo.T + bo
  k_wmma_gemm<1, 0><<<dim3(100, 1), 256, 0, stream>>>(xin, awv, abv, h1, 3200, 256, 256);
  k_wmma_gemm<1, 0><<<dim3(100, 1), 256, 0, stream>>>(h1, awo, abo, gen2, 3200, 256, 256);

  k_pack<<<6400, 256, 0, stream>>>(ve, gen2, xall);

  // Hoisted LSTM input projection for all 50 steps: XW = xall @ wih.T
  k_wmma_gemm<1, 0><<<dim3(200, 4), 256, 0, stream>>>(xall, wih, nullptr, XW, 6400, 1024, 256);

  k_zero<<<128, 256, 0, stream>>>(hcur, 32768);
  k_zero<<<128, 256, 0, stream>>>(ccur, 32768);
  for (int v = 0; v < VV; ++v) {
    k_wmma_gemm<1, 0><<<dim3(4, 4), 256, 0, stream>>>(hcur, whh, nullptr, gates, 128, 1024, 256);
    k_cell<<<128, 256, 0, stream>>>(XW, gates, bih, bhh, hcur, ccur, hsall, v);
  }

  // classifier: relu(hs @ c1w.T + c1b) -> relu(@ c2w.T + c2b) -> @ c3w.T + c3b
  k_wmma_gemm<1, 1><<<dim3(200, 4), 256, 0, stream>>>(hsall, c1w, c1b, XW, 6400, 1024, 256);
  k_wmma_gemm<1, 1><<<dim3(200, 2), 256, 0, stream>>>(XW, c2w, c2b, h2c, 6400, 512, 1024);
  k_clf3<<<50, 256, 0, stream>>>(h2c, c3w, c3b, out);
}